// TransformerCell_59115929862588
// MI455X (gfx1250) — compile-verified
//
#include <hip/hip_runtime.h>
#include <hip/hip_bf16.h>

#define BB 128
#define DD 1024
#define DSPLIT 4
#define DCH (DD / DSPLIT)   // 256

typedef float v2f __attribute__((ext_vector_type(2)));
typedef float v4f __attribute__((ext_vector_type(4)));
typedef float v8f __attribute__((ext_vector_type(8)));

// ---------------------------------------------------------------------------
// Kernel 1: q/k/v = phi(x @ W^T + b) via V_WMMA_F32_16X16X4_F32.
// One wave per 16x16 output tile. 3 matrices * (128/16)*(1024/16) = 1536 waves.
// A (x) lane layout: lane m=lane&15 holds row m, K pair selected by lane>>4.
// B (W^T) column n == W row n, K-contiguous in memory -> same load pattern.
// ---------------------------------------------------------------------------
__global__ void __launch_bounds__(256)
qkv_wmma_kernel(const float* __restrict__ x,
                const float* __restrict__ Wq, const float* __restrict__ bq,
                const float* __restrict__ Wk, const float* __restrict__ bk,
                const float* __restrict__ Wv, const float* __restrict__ bv,
                float* __restrict__ ws_qkv)   // [3][BB][DD]
{
    const int wid  = (blockIdx.x * blockDim.x + threadIdx.x) >> 5;
    const int lane = threadIdx.x & 31;
    if (wid >= 3 * (BB / 16) * (DD / 16)) return;

    const int mat  = wid / ((BB / 16) * (DD / 16));      // 0=q 1=k 2=v
    const int tile = wid % ((BB / 16) * (DD / 16));
    const int mt   = tile >> 6;                          // batch tile 0..7
    const int nt   = tile & 63;                          // out-col tile 0..63

    const float* W    = (mat == 0) ? Wq : (mat == 1) ? Wk : Wv;
    const float* bias = (mat == 0) ? bq : (mat == 1) ? bk : bv;
    float* outp = ws_qkv + (size_t)mat * BB * DD;

    const int m    = lane & 15;
    const int half = lane >> 4;

    // C init = bias broadcast: every C VGPR holds element n = nt*16 + (lane&15)
    const float bn = bias[nt * 16 + m];
    v8f c;
#pragma unroll
    for (int i = 0; i < 8; ++i) c[i] = bn;

    const float* arow = x + (size_t)(mt * 16 + m) * DD;  // x row (K contiguous)
    const float* brow = W + (size_t)(nt * 16 + m) * DD;  // W row n (K contiguous)

#pragma unroll 8
    for (int k0 = 0; k0 < DD; k0 += 4) {
        v2f a = *(const v2f*)(arow + k0 + 2 * half);     // K = k0+2*half, +1
        v2f b = *(const v2f*)(brow + k0 + 2 * half);
        c = __builtin_amdgcn_wmma_f32_16x16x4_f32(
                /*neg_a=*/false, a, /*neg_b=*/false, b,
                /*c_mod=*/(short)0, c, /*reuse_a=*/false, /*reuse_b=*/false);
    }

    // Epilogue: phi(u)=elu(u)+1 for q,k (u>0 -> u+1 ; else exp(u)); identity for v.
    // C layout: VGPR g -> row (g + 8*half), col = nt*16 + (lane&15).
#pragma unroll
    for (int g = 0; g < 8; ++g) {
        float u = c[g];
        if (mat < 2) u = (u > 0.0f) ? (u + 1.0f) : __expf(u);
        const int row = mt * 16 + g + 8 * half;
        outp[(size_t)row * DD + nt * 16 + m] = u;
    }
}

// ---------------------------------------------------------------------------
// Kernel 2: z_new = z + k ; denom[b] = sum_d q*z_new. One block per batch.
// ---------------------------------------------------------------------------
__global__ void __launch_bounds__(256)
zdenom_kernel(const float* __restrict__ z,
              const float* __restrict__ q_ws,
              const float* __restrict__ k_ws,
              float* __restrict__ z_out,      // d_out z_new section [BB][DD]
              float* __restrict__ denom)      // [BB]
{
    const int b = blockIdx.x;
    const int t = threadIdx.x;
    float part = 0.0f;
    for (int d = t; d < DD; d += 256) {
        const float zn = z[(size_t)b * DD + d] + k_ws[(size_t)b * DD + d];
        z_out[(size_t)b * DD + d] = zn;
        part += q_ws[(size_t)b * DD + d] * zn;
    }
    __shared__ float red[256];
    red[t] = part;
    __syncthreads();
#pragma unroll
    for (int off = 128; off > 0; off >>= 1) {
        if (t < off) red[t] += red[t + off];
        __syncthreads();
    }
    if (t == 0) denom[b] = red[0];
}

// ---------------------------------------------------------------------------
// Kernel 3: fused state update + numerator (single pass over 512 MB state).
//   s_new[b,d,e] = s[b,d,e] + k[b,d]*v[b,e]
//   num[b,e]    += sum_d q[b,d]*s_new[b,d,e]
// grid = BB*DSPLIT blocks, 256 threads, v4f per thread over e (full 1024).
// Non-temporal on the s stream (512 MB >> 192 MB L2).
// ---------------------------------------------------------------------------
__global__ void __launch_bounds__(256)
state_stream_kernel(const float* __restrict__ s,
                    const float* __restrict__ q_ws,
                    const float* __restrict__ k_ws,
                    const float* __restrict__ v_ws,
                    float* __restrict__ s_new,  // d_out s section
                    float* __restrict__ num)    // [BB][DD] accumulator
{
    const int b  = blockIdx.x / DSPLIT;
    const int dc = blockIdx.x % DSPLIT;
    const int t  = threadIdx.x;
    const int e0 = t * 4;

    const v4f v4 = *(const v4f*)(v_ws + (size_t)b * DD + e0);

    __shared__ float kq[2][DCH];
    kq[0][t] = k_ws[(size_t)b * DD + dc * DCH + t];
    kq[1][t] = q_ws[(size_t)b * DD + dc * DCH + t];
    __syncthreads();

    const size_t base = (size_t)b * DD * DD + (size_t)(dc * DCH) * DD + e0;
    const float* sp = s + base;
    float*       dp = s_new + base;

    v4f acc = {0.f, 0.f, 0.f, 0.f};
#pragma unroll 4
    for (int i = 0; i < DCH; ++i) {
        const float kd = kq[0][i];
        const float qd = kq[1][i];
        v4f sv = __builtin_nontemporal_load((const v4f*)(sp + (size_t)i * DD));
#pragma unroll
        for (int j = 0; j < 4; ++j) sv[j] = fmaf(kd, v4[j], sv[j]);
        __builtin_nontemporal_store(sv, (v4f*)(dp + (size_t)i * DD));
#pragma unroll
        for (int j = 0; j < 4; ++j) acc[j] = fmaf(qd, sv[j], acc[j]);
    }

    float* np = num + (size_t)b * DD + e0;
    atomicAdd(np + 0, acc[0]);
    atomicAdd(np + 1, acc[1]);
    atomicAdd(np + 2, acc[2]);
    atomicAdd(np + 3, acc[3]);
}

// ---------------------------------------------------------------------------
// Kernel 4: out = relu(num/denom + x)
// ---------------------------------------------------------------------------
__global__ void __launch_bounds__(256)
finalize_kernel(const float* __restrict__ x,
                const float* __restrict__ num,
                const float* __restrict__ denom,
                float* __restrict__ out)
{
    const int i = blockIdx.x * 256 + threadIdx.x;
    if (i >= BB * DD) return;
    const int b = i / DD;
    const float r = num[i] / denom[b] + x[i];
    out[i] = (r > 0.0f) ? r : 0.0f;
}

// ---------------------------------------------------------------------------
extern "C" void kernel_launch(void* const* d_in, const int* in_sizes, int n_in,
                              void* d_out, int out_size, void* d_ws, size_t ws_size,
                              hipStream_t stream) {
    const float* x  = (const float*)d_in[0];
    const float* s  = (const float*)d_in[1];
    const float* z  = (const float*)d_in[2];
    const float* Wq = (const float*)d_in[3];
    const float* bq = (const float*)d_in[4];
    const float* Wk = (const float*)d_in[5];
    const float* bk = (const float*)d_in[6];
    const float* Wv = (const float*)d_in[7];
    const float* bv = (const float*)d_in[8];

    float* out = (float*)d_out;
    // d_out layout: out [BB*DD] | s_new [BB*DD*DD] | z_new [BB*DD]
    float* out_o = out;
    float* s_new = out + (size_t)BB * DD;
    float* z_new = out + (size_t)BB * DD + (size_t)BB * DD * DD;

    // workspace: q | k | v | num  (BB*DD each)  | denom (BB)
    float* ws    = (float*)d_ws;
    float* q_ws  = ws;
    float* k_ws  = ws + 1 * (size_t)BB * DD;
    float* v_ws  = ws + 2 * (size_t)BB * DD;
    float* num   = ws + 3 * (size_t)BB * DD;
    float* denom = ws + 4 * (size_t)BB * DD;

    // 1) QKV GEMMs on WMMA: 1536 waves -> 192 blocks of 256 threads
    qkv_wmma_kernel<<<192, 256, 0, stream>>>(x, Wq, bq, Wk, bk, Wv, bv, q_ws);

    // 2) zero the numerator accumulator
    (void)hipMemsetAsync(num, 0, (size_t)BB * DD * sizeof(float), stream);

    // 3) z_new + denom
    zdenom_kernel<<<BB, 256, 0, stream>>>(z, q_ws, k_ws, z_new, denom);

    // 4) fused 1-pass state stream (the 1 GB bandwidth kernel)
    state_stream_kernel<<<BB * DSPLIT, 256, 0, stream>>>(s, q_ws, k_ws, v_ws,
                                                         s_new, num);

    // 5) epilogue
    finalize_kernel<<<(BB * DD + 255) / 256, 256, 0, stream>>>(x, num, denom, out_o);
}